// ComplexHST_47596827574749
// MI455X (gfx1250) — compile-verified
//
#include <hip/hip_runtime.h>
#include <cmath>

typedef float v2f __attribute__((ext_vector_type(2)));
typedef float v8f __attribute__((ext_vector_type(8)));

// ---------------------------------------------------------------------------
// rect(): R(z) = i*log(-i*(w + sign(Im z)*sqrt(w-1)*sqrt(w+1))), w = 2z/pi.
// Output: Re = -arg(r0), Im = log|r0|.
// ---------------------------------------------------------------------------
__device__ __forceinline__ void csqrt_f32(float a, float b, float& re, float& im) {
    float r = sqrtf(fmaf(a, a, b * b));
    re = sqrtf(fmaxf(0.0f, 0.5f * (r + a)));
    im = copysignf(sqrtf(fmaxf(0.0f, 0.5f * (r - a))), b);
}

__device__ __forceinline__ void rect_f32(float a, float b, float& ore, float& oim) {
    const float C = 0.63661977236758134f;     // 2/pi
    float wr = a * C, wi = b * C;
    float s1r, s1i, s2r, s2i;
    csqrt_f32(wr - 1.0f, wi, s1r, s1i);
    csqrt_f32(wr + 1.0f, wi, s2r, s2i);
    float sr = s1r * s2r - s1i * s2i;
    float si = s1r * s2i + s1i * s2r;
    float sg = (b < 0.0f) ? -1.0f : 1.0f;     // sign(0) -> 1 as in reference
    float qr = wr + sg * sr;
    float qi = wi + sg * si;
    float rr = qi, ri = -qr;                  // r0 = -i*q
    ore = -atan2f(ri, rr);
    oim = 0.5f * logf(fmaf(rr, rr, ri * ri));
}

// Initial transform: u1 = rect(f + 0i) -> separate re/im planes.
__global__ void rect0_kernel(const float* __restrict__ f,
                             float* __restrict__ out_re,
                             float* __restrict__ out_im, int n) {
    int i = blockIdx.x * blockDim.x + threadIdx.x;
    if (i >= n) return;
    float ore, oim;
    rect_f32(f[i], 0.0f, ore, oim);
    out_re[i] = ore;
    out_im[i] = oim;
}

// ---------------------------------------------------------------------------
// One DWT level via V_WMMA_F32_16X16X4_F32, with rect() fused on cA.
// Tile = 16 batch rows (M-dim) x 16 output positions (N-dim).
//   D[m][n] = sum_j A[m][j] * B[j][n],  j in [0,48)  (12 K=4 steps)
//   A[m][j] = x[r0+m][(2*i0 - 16 + j) mod N]   (even window start ->
//             per-lane K-pairs are 8B aligned -> global_load_b64 fast path;
//             wrap only possible for the i0==0 tile)
//   B[j][n] = filt[2n + 16 - j]   (Toeplitz filter bank in LDS; j=0 col = 0)
// cD -> interleaved complex into d_out slice (coalesced along lanes);
// rect(cA) -> re/im planes for the next level, or interleaved (final level).
// ---------------------------------------------------------------------------
__global__ __launch_bounds__(256) void dwt_wmma_kernel(
        const float* __restrict__ in_re, const float* __restrict__ in_im,
        const float* __restrict__ lo,    const float* __restrict__ hi,
        float* __restrict__ cd_out,      // interleaved complex
        float* __restrict__ ua_re,       float* __restrict__ ua_im,
        float* __restrict__ ua_ilv,      int final_ilv,
        int N) {
    const int M = N >> 1;
    __shared__ __align__(16) float sB[2][12][32][2];   // [filter][kstep][lane][pair]

    const int tid = threadIdx.x;
    if (tid < 32) {
        int n  = tid & 15;
        int kh = (tid >> 4) * 2;
        for (int t = 0; t < 12; ++t) {
            for (int e = 0; e < 2; ++e) {
                int j = 4 * t + kh + e;
                int k = 2 * n + 16 - j;
                float vl = 0.0f, vh = 0.0f;
                if (k >= 0 && k < 16) { vl = lo[k]; vh = hi[k]; }
                sB[0][t][tid][e] = vl;
                sB[1][t][tid][e] = vh;
            }
        }
    }
    __syncthreads();

    const int lane    = tid & 31;
    const int wave_id = blockIdx.x * (blockDim.x >> 5) + (tid >> 5);
    const int n_tiles = 2 * (M >> 4);
    if (wave_id >= n_tiles) return;          // wave-uniform (EXEC stays all-ones)

    const int r0    = (wave_id & 1) << 4;    // row group: 0 or 16
    const int i0    = (wave_id >> 1) << 4;   // 16 output positions
    const int lanem = lane & 15;
    const int half  = lane >> 4;
    const int kh    = half * 2;
    const int row   = r0 + lanem;            // A-matrix: lane = M (row)
    const int base  = 2 * i0 - 16;           // even window start

    v8f accA0 = {}, accA1 = {}, accD0 = {}, accD1 = {};

    auto run_component = [&](const float* __restrict__ src, v8f& aA, v8f& aD) {
        const float* rp = src + (size_t)row * N;
        v2f areg[12];
        if (base >= 0) {
            // fast path: contiguous, 8B-aligned pair loads at immediate offsets
            const v2f* wp = reinterpret_cast<const v2f*>(rp + base + kh);
#pragma unroll
            for (int t = 0; t < 12; ++t) areg[t] = wp[2 * t];
        } else {
            // wrap path: only the i0==0 tile
#pragma unroll
            for (int t = 0; t < 12; ++t) {
                int j0 = 4 * t + kh;
                int x0 = base + j0;     if (x0 < 0) x0 += N;
                int x1 = base + j0 + 1; if (x1 < 0) x1 += N;
                v2f v; v[0] = rp[x0]; v[1] = rp[x1];
                areg[t] = v;
            }
        }
#pragma unroll
        for (int t = 0; t < 12; ++t) {
            v2f bl = *reinterpret_cast<const v2f*>(&sB[0][t][lane][0]);
            v2f bh = *reinterpret_cast<const v2f*>(&sB[1][t][lane][0]);
            aA = __builtin_amdgcn_wmma_f32_16x16x4_f32(false, areg[t], false, bl,
                                                       (short)0, aA, false, false);
            aD = __builtin_amdgcn_wmma_f32_16x16x4_f32(false, areg[t], false, bh,
                                                       (short)0, aD, false, false);
        }
    };

    run_component(in_re, accA0, accD0);   // real component
    run_component(in_im, accA1, accD1);   // imag component

    // D layout: VGPR g -> M = g (lanes 0-15) / g+8 (lanes 16-31); N = lane%16
#pragma unroll
    for (int g = 0; g < 8; ++g) {
        int rout = r0 + g + (half ? 8 : 0);
        size_t o = (size_t)rout * M + (size_t)i0 + lanem;
        reinterpret_cast<float2*>(cd_out)[o] = make_float2(accD0[g], accD1[g]);
        float ur, ui;
        rect_f32(accA0[g], accA1[g], ur, ui);           // fused rect(cA)
        if (final_ilv) {
            reinterpret_cast<float2*>(ua_ilv)[o] = make_float2(ur, ui);
        } else {
            ua_re[o] = ur;
            ua_im[o] = ui;
        }
    }
}

// ---------------------------------------------------------------------------
// Host-side orchestration: rect0 -> 4x fused (DWT level + rect).
// ---------------------------------------------------------------------------
extern "C" void kernel_launch(void* const* d_in, const int* in_sizes, int n_in,
                              void* d_out, int out_size, void* d_ws, size_t ws_size,
                              hipStream_t stream) {
    const float* f  = (const float*)d_in[0];
    const float* lo = (const float*)d_in[1];
    const float* hi = (const float*)d_in[2];
    float* out = (float*)d_out;

    const int total0 = in_sizes[0];      // 32 * N
    const int N      = total0 / 32;

    char* ws = (char*)d_ws;
    const size_t p = (size_t)total0 * sizeof(float);   // one full-size plane
    float* b0r = (float*)(ws);
    float* b0i = (float*)(ws + p);
    float* b1r = (float*)(ws + 2 * p);
    float* b1i = (float*)(ws + 2 * p + p / 2);

    // output offsets (complex elements): cD1,cD2,cD3,cD4,u_final
    const size_t c1 = (size_t)32 * (N / 2);
    const size_t c2 = (size_t)32 * (N / 4);
    const size_t c3 = (size_t)32 * (N / 8);
    const size_t c4 = (size_t)32 * (N / 16);
    float* outD1 = out;
    float* outD2 = out + 2 * c1;
    float* outD3 = out + 2 * (c1 + c2);
    float* outD4 = out + 2 * (c1 + c2 + c3);
    float* outU  = out + 2 * (c1 + c2 + c3 + c4);

    const dim3 blk(256);
    // conv grid: 2*(M/16) tiles, 8 waves/block -> M/64 blocks
    auto cblocks = [](int M) { return dim3((unsigned)(M / 64)); };

    // u1 = rect(f + 0i) -> planes b0
    rect0_kernel<<<dim3((unsigned)((total0 + 255) / 256)), blk, 0, stream>>>(
        f, b0r, b0i, total0);

    // level 1: b0 (N)   -> cD1, u2 = rect(cA1) planes b1
    dwt_wmma_kernel<<<cblocks(N / 2), blk, 0, stream>>>(
        b0r, b0i, lo, hi, outD1, b1r, b1i, (float*)nullptr, 0, N);
    // level 2: b1 (N/2) -> cD2, u3 planes b0
    dwt_wmma_kernel<<<cblocks(N / 4), blk, 0, stream>>>(
        b1r, b1i, lo, hi, outD2, b0r, b0i, (float*)nullptr, 0, N / 2);
    // level 3: b0 (N/4) -> cD3, u4 planes b1
    dwt_wmma_kernel<<<cblocks(N / 8), blk, 0, stream>>>(
        b0r, b0i, lo, hi, outD3, b1r, b1i, (float*)nullptr, 0, N / 4);
    // level 4: b1 (N/8) -> cD4, u_final interleaved into d_out tail
    dwt_wmma_kernel<<<cblocks(N / 16), blk, 0, stream>>>(
        b1r, b1i, lo, hi, outD4, (float*)nullptr, (float*)nullptr, outU, 1, N / 8);
}